// SelfTrans_23811298689071
// MI455X (gfx1250) — compile-verified
//
#include <hip/hip_runtime.h>
#include <math.h>

// ---------------------------------------------------------------- constants
#define BB      4
#define CH      512
#define HH      40
#define WW      40
#define NSP     1600          // 40*40
#define H2      20
#define W2      20
#define NSP2    400           // 20*20
#define NHEAD   8
#define DK      64
#define DV      64
#define NMIX    2
#define DSUB    32            // DK / NMIX
#define BP      32            // B * NHEAD
#define TEMP_INV (0.125f)     // 1/sqrt(64)

typedef float v2f __attribute__((ext_vector_type(2)));
typedef float v8f __attribute__((ext_vector_type(8)));
typedef int   v4i_t __attribute__((ext_vector_type(4)));

// gfx1250 async global->LDS copy, guarded so the file compiles on any toolchain
#if defined(__has_builtin)
#if __has_builtin(__builtin_amdgcn_global_load_async_to_lds_b128) && \
    __has_builtin(__builtin_amdgcn_s_wait_asynccnt)
#define USE_ASYNC_LDS 1
#endif
#endif
#ifndef USE_ASYNC_LDS
#define USE_ASYNC_LDS 0
#endif

#if USE_ASYNC_LDS
__device__ __forceinline__ void async_copy16(const float* g, float* l) {
    // builtin signature (probe-verified): (int4* global_src, int4* lds_dst, imm offset, imm cpol)
    __builtin_amdgcn_global_load_async_to_lds_b128((v4i_t*)g, (v4i_t*)l, 0, 0);
}
#endif

// ---------------------------------------------------------------- avg pool 3x3 s2 p1 (count_include_pad=False)
__global__ __launch_bounds__(256) void avgpool_kernel(const float* __restrict__ x,
                                                      float* __restrict__ xp) {
    int idx = blockIdx.x * blockDim.x + threadIdx.x;
    if (idx >= BB * CH * H2 * W2) return;
    int ow = idx % W2;
    int oh = (idx / W2) % H2;
    int bc = idx / (H2 * W2);
    const float* src = x + (long long)bc * NSP;
    int h0 = 2 * oh - 1, w0 = 2 * ow - 1;
    float s = 0.f; int cnt = 0;
    #pragma unroll
    for (int dh = 0; dh < 3; ++dh) {
        int ih = h0 + dh;
        if (ih < 0 || ih >= HH) continue;
        #pragma unroll
        for (int dw = 0; dw < 3; ++dw) {
            int iw = w0 + dw;
            if (iw < 0 || iw >= WW) continue;
            s += src[ih * WW + iw]; ++cnt;
        }
    }
    xp[idx] = s / (float)cnt;
}

// ---------------------------------------------------------------- fp32 WMMA GEMM, async-LDS double-buffered
// C[z] = Wm[z] (MxK row-major) @ Xm[z] (KxN row-major) + bias
// 8 waves/WG arranged 4(rows) x 2(cols). Wave tile 16x64 (4 accums). WG tile 64x128.
// K-slab = 16, double buffered: async-copy slab i+1 while WMMAs consume slab i.
// Requirements (all call sites satisfy): K % 16 == 0, M % 64 == 0, N % 8 == 0, N >= 8.
#define SB_STRIDE 144   // 128 + 16 pad: lane halves hit disjoint 16-aligned bank groups
#define SA_STRIDE 20    // 16 + 4 pad: 16B-aligned async dst, 8B-aligned b64 frag, conflict-free

__global__ __launch_bounds__(256) void gemm_wmma_f32(
    const float* __restrict__ Wm, const float* __restrict__ Xm,
    const float* __restrict__ bias, float* __restrict__ Cm,
    int M, int K, int N,
    long long sW, long long sX, long long sC)
{
    __shared__ float sA[2][64 * SA_STRIDE];
    __shared__ float sB[2][16 * SB_STRIDE];

    const int tid   = threadIdx.x;
    const int lane  = tid & 31;
    const int wave  = tid >> 5;
    const int wr    = wave >> 1;             // 0..3
    const int wc    = wave & 1;              // 0..1
    const int lmod  = lane & 15;
    const int khalf = lane >> 4;             // 0 or 1 (upper lane half)
    const int m0blk = blockIdx.y * 64;
    const int nblk  = blockIdx.x * 128;

    const float* Wp = Wm + (long long)blockIdx.z * sW;
    const float* Xp = Xm + (long long)blockIdx.z * sX;
    float*       Cp = Cm + (long long)blockIdx.z * sC;

    // cooperative-load coordinates
    const int ar = tid >> 2;                 // 0..63  : A row
    const int ks = (tid & 3) * 4;            // 0,4,8,12: A k-segment
    const int kr = tid >> 4;                 // 0..15  : B row
    const int cs = (tid & 15) * 8;           // 0..120 : B col-segment
    int gb = nblk + cs;                      // clamp: OOB cols only feed unstored outputs
    if (gb + 8 > N) gb = N - 8;

    auto issue = [&](int k0, int bufsel) {
        const float* ag = Wp + (long long)(m0blk + ar) * K + (k0 + ks);
        const float* bg = Xp + (long long)(k0 + kr) * N + gb;
        float* al = &sA[bufsel][ar * SA_STRIDE + ks];
        float* bl = &sB[bufsel][kr * SB_STRIDE + cs];
#if USE_ASYNC_LDS
        async_copy16(ag, al);
        async_copy16(bg, bl);
        async_copy16(bg + 4, bl + 4);
#else
        float4 av = *(const float4*)ag;
        al[0] = av.x; al[1] = av.y; al[2] = av.z; al[3] = av.w;
        float4 bv0 = *(const float4*)bg;
        float4 bv1 = *(const float4*)(bg + 4);
        bl[0] = bv0.x; bl[1] = bv0.y; bl[2] = bv0.z; bl[3] = bv0.w;
        bl[4] = bv1.x; bl[5] = bv1.y; bl[6] = bv1.z; bl[7] = bv1.w;
#endif
        if (k0 + 16 < K) {
            // warm caches for the slab after this one (global_prefetch_b8)
            __builtin_prefetch(ag + 16, 0, 1);
            __builtin_prefetch(bg + (long long)16 * N, 0, 1);
        }
    };

    v8f acc[4] = {};

    issue(0, 0);
#if USE_ASYNC_LDS
    __builtin_amdgcn_s_wait_asynccnt(0);
#endif
    __syncthreads();

    const int nslab = K >> 4;
    for (int i = 0; i < nslab; ++i) {
        const int cur = i & 1;
        if (i + 1 < nslab) issue((i + 1) << 4, cur ^ 1);

        const float* arow  = &sA[cur][(wr * 16 + lmod) * SA_STRIDE];
        const float* bbase = &sB[cur][0];
        #pragma unroll
        for (int kk = 0; kk < 16; kk += 4) {
            v2f a = *(const v2f*)&arow[kk + 2 * khalf];   // 8B-aligned ds_load_b64
            #pragma unroll
            for (int f = 0; f < 4; ++f) {
                const int col = wc * 64 + 16 * f + lmod;
                v2f b;
                b.x = bbase[(kk + khalf) * SB_STRIDE + col];
                b.y = bbase[(kk + khalf + 2) * SB_STRIDE + col];
                acc[f] = __builtin_amdgcn_wmma_f32_16x16x4_f32(
                    false, a, false, b, (short)0, acc[f], false, false);
            }
        }
#if USE_ASYNC_LDS
        __builtin_amdgcn_s_wait_asynccnt(0);
#endif
        __syncthreads();
    }

    // C/D layout: VGPR r -> (M = m0+r) lanes 0-15, (M = m0+r+8) lanes 16-31
    const int m0 = m0blk + wr * 16;
    #pragma unroll
    for (int f = 0; f < 4; ++f) {
        const int cc = nblk + wc * 64 + 16 * f + lmod;
        #pragma unroll
        for (int r = 0; r < 8; ++r) {
            const int rr = m0 + r + khalf * 8;
            if (cc < N) {
                float v = acc[f][r];
                if (bias) v += bias[rr];
                Cp[(long long)rr * N + cc] = v;
            }
        }
    }
}

// ---------------------------------------------------------------- mean(kt) over N, then mixture softmax -> pi
__global__ __launch_bounds__(256) void barkt_pi_kernel(
    const float* __restrict__ kt, const float* __restrict__ mixw,
    float* __restrict__ pi)
{
    const int bp  = blockIdx.x;   // 0..31
    const int tid = threadIdx.x;
    __shared__ float sred[256];
    __shared__ float sbar[DK];
    const int r    = tid >> 2;    // 0..63 (row)
    const int part = tid & 3;     // 4-way split of 1600
    const float* src = kt + (long long)bp * DK * NSP + (long long)r * NSP;
    float s = 0.f;
    for (int n = part * 400; n < part * 400 + 400; ++n) s += src[n];
    sred[tid] = s;
    __syncthreads();
    if (part == 0) {
        float t = sred[tid] + sred[tid + 1] + sred[tid + 2] + sred[tid + 3];
        sbar[r] = t * (1.0f / (float)NSP);
    }
    __syncthreads();
    if (tid == 0) {
        float l0 = 0.f, l1 = 0.f;
        for (int d = 0; d < DK; ++d) {
            l0 += mixw[d]      * sbar[d];
            l1 += mixw[DK + d] * sbar[d];
        }
        float mx = fmaxf(l0, l1);
        float e0 = __expf(l0 - mx), e1 = __expf(l1 - mx);
        float inv = 1.0f / (e0 + e1);
        pi[bp * NMIX + 0] = e0 * inv;
        pi[bp * NMIX + 1] = e1 * inv;
    }
}

// ---------------------------------------------------------------- in-place: kt -> pi * softmax_d(kt) / temp
__global__ __launch_bounds__(256) void softmax_scale_kernel(
    float* __restrict__ kt, const float* __restrict__ pi)
{
    int idx = blockIdx.x * blockDim.x + threadIdx.x;
    if (idx >= BP * NMIX * NSP) return;
    const int n  = idx % NSP;
    const int m  = (idx / NSP) & (NMIX - 1);
    const int bp = idx / (NMIX * NSP);
    float* col = kt + (long long)bp * DK * NSP + (long long)m * DSUB * NSP + n;
    float mx = -INFINITY;
    #pragma unroll
    for (int d = 0; d < DSUB; ++d) mx = fmaxf(mx, col[d * NSP]);
    float vals[DSUB];
    float ssum = 0.f;
    #pragma unroll
    for (int d = 0; d < DSUB; ++d) {
        float e = __expf(col[d * NSP] - mx);
        vals[d] = e; ssum += e;
    }
    const float scale = pi[bp * NMIX + m] * TEMP_INV / ssum;
    #pragma unroll
    for (int d = 0; d < DSUB; ++d) col[d * NSP] = vals[d] * scale;
}

// ---------------------------------------------------------------- MT[b'][dv][md] = sum_k vt[b'][md][k] * qt[b'][dv][k]
__global__ __launch_bounds__(256) void vtq_kernel(
    const float* __restrict__ vt, const float* __restrict__ qt, float* __restrict__ MT)
{
    int idx = blockIdx.x * blockDim.x + threadIdx.x;
    if (idx >= BP * DV * DK) return;
    const int md = idx & 63;
    const int dv = (idx >> 6) & 63;
    const int bp = idx >> 12;
    const float* v = vt + (long long)bp * DK * NSP2 + md * NSP2;
    const float* q = qt + (long long)bp * DK * NSP2 + dv * NSP2;
    float s = 0.f;
    for (int k = 0; k < NSP2; ++k) s = fmaf(v[k], q[k], s);
    MT[idx] = s;   // idx = bp*4096 + dv*64 + md
}

// ---------------------------------------------------------------- BN batch-stats: per-channel sum / sumsq
__global__ __launch_bounds__(256) void bn_reduce_kernel(
    const float* __restrict__ y, float* __restrict__ bnsum, float* __restrict__ bnsq)
{
    const int c   = blockIdx.x;   // 0..511
    const int tid = threadIdx.x;
    float s = 0.f, s2 = 0.f;
    for (int b = 0; b < BB; ++b) {
        const float* p = y + ((long long)b * CH + c) * NSP;
        for (int n = tid; n < NSP; n += 256) {
            float v = p[n];
            s += v; s2 = fmaf(v, v, s2);
        }
    }
    __shared__ float sh[256], sh2[256];
    sh[tid] = s; sh2[tid] = s2;
    __syncthreads();
    for (int off = 128; off > 0; off >>= 1) {
        if (tid < off) { sh[tid] += sh[tid + off]; sh2[tid] += sh2[tid + off]; }
        __syncthreads();
    }
    if (tid == 0) { bnsum[c] = sh[0]; bnsq[c] = sh2[0]; }
}

// ---------------------------------------------------------------- BN affine + residual (in-place on d_out)
__global__ __launch_bounds__(256) void bn_apply_kernel(
    float* __restrict__ y, const float* __restrict__ x,
    const float* __restrict__ bnsum, const float* __restrict__ bnsq,
    const float* __restrict__ gamma, const float* __restrict__ beta)
{
    int idx = blockIdx.x * blockDim.x + threadIdx.x;
    if (idx >= BB * CH * NSP) return;
    const int c = (idx / NSP) & (CH - 1);
    const float cntinv = 1.0f / (float)(BB * NSP);
    const float mean = bnsum[c] * cntinv;
    const float var  = bnsq[c] * cntinv - mean * mean;
    const float inv  = rsqrtf(var + 1e-5f);
    y[idx] = (y[idx] - mean) * inv * gamma[c] + beta[c] + x[idx];
}

// ---------------------------------------------------------------- launcher
extern "C" void kernel_launch(void* const* d_in, const int* in_sizes, int n_in,
                              void* d_out, int out_size, void* d_ws, size_t ws_size,
                              hipStream_t stream) {
    const float* x     = (const float*)d_in[0];
    const float* w_qk  = (const float*)d_in[1];
    const float* b_qk  = (const float*)d_in[2];
    const float* w_v   = (const float*)d_in[3];
    const float* b_v   = (const float*)d_in[4];
    const float* mixw  = (const float*)d_in[5];
    const float* w_out = (const float*)d_in[6];
    const float* gamma = (const float*)d_in[7];
    const float* beta  = (const float*)d_in[8];
    float* y  = (float*)d_out;
    float* ws = (float*)d_ws;

    // workspace layout (floats)
    float* xp    = ws;                               // 819200
    float* kt    = xp  + (long long)BB * CH * NSP2;  // 3276800 (becomes A in-place)
    float* qt    = kt  + (long long)BB * CH * NSP;   // 819200
    float* vt    = qt  + (long long)BB * CH * NSP2;  // 819200
    float* MT    = vt  + (long long)BB * CH * NSP2;  // 131072
    float* OUT   = MT  + (long long)BP * DV * DK;    // 3276800
    float* pi    = OUT + (long long)BB * CH * NSP;   // 64
    float* bnsum = pi  + BP * NMIX;                  // 512
    float* bnsq  = bnsum + CH;                       // 512

    // 1) avg pool
    avgpool_kernel<<<(BB * CH * NSP2 + 255) / 256, 256, 0, stream>>>(x, xp);

    // 2) kt = w_qk @ x + b_qk   (per batch: 512x512x1600)
    {
        dim3 g((NSP + 127) / 128, CH / 64, BB);
        gemm_wmma_f32<<<g, 256, 0, stream>>>(w_qk, x, b_qk, kt, CH, CH, NSP,
                                             0LL, (long long)CH * NSP, (long long)CH * NSP);
    }
    // 3) qt, vt from pooled xp  (per batch: 512x512x400)
    {
        dim3 g((NSP2 + 127) / 128, CH / 64, BB);
        gemm_wmma_f32<<<g, 256, 0, stream>>>(w_qk, xp, b_qk, qt, CH, CH, NSP2,
                                             0LL, (long long)CH * NSP2, (long long)CH * NSP2);
        gemm_wmma_f32<<<g, 256, 0, stream>>>(w_v, xp, b_v, vt, CH, CH, NSP2,
                                             0LL, (long long)CH * NSP2, (long long)CH * NSP2);
    }
    // 4) mixture weights pi
    barkt_pi_kernel<<<BP, 256, 0, stream>>>(kt, mixw, pi);

    // 5) kt -> pi * softmax_d(kt) / temperature (in-place)
    softmax_scale_kernel<<<(BP * NMIX * NSP + 255) / 256, 256, 0, stream>>>(kt, pi);

    // 6) MT = Vt @ Q^T  (tiny 64x64 per b')
    vtq_kernel<<<(BP * DV * DK + 255) / 256, 256, 0, stream>>>(vt, qt, MT);

    // 7) OUT = MT @ A   (per b': 64x64x1600) -> lands as [b][c][n]
    {
        dim3 g((NSP + 127) / 128, 1, BP);
        gemm_wmma_f32<<<g, 256, 0, stream>>>(MT, kt, nullptr, OUT, DV, DK, NSP,
                                             (long long)DV * DK,
                                             (long long)DK * NSP,
                                             (long long)DV * NSP);
    }
    // 8) y = w_out @ OUT (per batch: 512x512x1600) into d_out
    {
        dim3 g((NSP + 127) / 128, CH / 64, BB);
        gemm_wmma_f32<<<g, 256, 0, stream>>>(w_out, OUT, nullptr, y, CH, CH, NSP,
                                             0LL, (long long)CH * NSP, (long long)CH * NSP);
    }
    // 9) batch-norm statistics + apply + residual
    bn_reduce_kernel<<<CH, 256, 0, stream>>>(y, bnsum, bnsq);
    bn_apply_kernel<<<(BB * CH * NSP + 255) / 256, 256, 0, stream>>>(y, x, bnsum, bnsq, gamma, beta);
}